// AdaGAE_1030792151698
// MI455X (gfx1250) — compile-verified
//
#include <hip/hip_runtime.h>

typedef __attribute__((ext_vector_type(2))) float v2f;
typedef __attribute__((ext_vector_type(8))) float v8f;

#define Nn   8192
#define DIN  32
#define DMID 24
#define DEMB 12
#define KCH  2048   // K-chunk for A-GEMM split (4 chunks)
#define JCH  1024   // j-chunk for softmax split (8 chunks)
#define ST1  132    // LDS column stride (floats) for gemm B-panels: 132 mod 64 = 4 -> conflict-free
#define SE   18     // LDS row stride for softmax emb panel

static __device__ __forceinline__ v8f wmma4(v2f a, v2f b, v8f c) {
  // D = A(16x4,f32) * B(4x16,f32) + C(16x16,f32)
  return __builtin_amdgcn_wmma_f32_16x16x4_f32(false, a, false, b, (short)0, c, false, false);
}

// ---------- T1 = X @ W1  [8192,32]x[32,24] ----------
__global__ __launch_bounds__(256) void k_xw1(const float* __restrict__ X,
                                             const float* __restrict__ W1,
                                             float* __restrict__ T1) {
  int i = blockIdx.x * 256 + threadIdx.x;
  if (i >= Nn) return;
  float acc[DMID];
#pragma unroll
  for (int n = 0; n < DMID; ++n) acc[n] = 0.f;
#pragma unroll 4
  for (int k = 0; k < DIN; ++k) {
    float xk = X[(size_t)i * DIN + k];
#pragma unroll
    for (int n = 0; n < DMID; ++n) acc[n] = fmaf(xk, W1[k * DMID + n], acc[n]);
  }
#pragma unroll
  for (int n = 0; n < DMID; ++n) T1[(size_t)i * DMID + n] = acc[n];
}

// ---------- hpart[c] = A[:, kchunk] @ T1[kchunk, :]  (WMMA f32) ----------
// B panel staged COLUMN-MAJOR in LDS: bs[col*ST1 + krow] -> each lane's B-frag
// (two consecutive K rows, fixed col) is one aligned ds_load_b64, no repacking.
__global__ __launch_bounds__(256) void k_gemm1(const float* __restrict__ A,
                                               const float* __restrict__ T1,
                                               float* __restrict__ hpart) {
  __shared__ float bs[32 * ST1];             // 16.5 KB; cols 24..31 pre-zeroed
  const int lane = threadIdx.x & 31;
  const int wave = threadIdx.x >> 5;
  const int half = lane >> 4;                // wave32: lanes 0-15 / 16-31
  const int n    = lane & 15;
  const int i0   = blockIdx.x * 128 + wave * 16;
  const int kbase = blockIdx.y * KCH;

  // zero pad columns once (copy loop below never touches them)
  for (int t = threadIdx.x; t < 8 * 128; t += 256)
    bs[(24 + (t >> 7)) * ST1 + (t & 127)] = 0.f;
  __syncthreads();

  v8f acc0 = {0,0,0,0,0,0,0,0};
  v8f acc1 = {0,0,0,0,0,0,0,0};
  const float* aptr = A + (size_t)(i0 + n) * Nn + kbase + 2 * half;

  for (int kt = 0; kt < KCH; kt += 128) {
    for (int t = threadIdx.x; t < 128 * DMID; t += 256) {
      int row = t / DMID, col = t - row * DMID;
      bs[col * ST1 + row] = T1[(size_t)(kbase + kt) * DMID + t];  // coalesced global read
    }
    __syncthreads();
#pragma unroll 4
    for (int kk = 0; kk < 128; kk += 4) {
      v2f a = *(const v2f*)(aptr + kt + kk);        // A frag: M=n, K=2*half+{0,1}
      const int kr = kk + 2 * half;
      v2f b0 = *(const v2f*)(bs + n * ST1 + kr);          // cols 0..15
      v2f b1 = *(const v2f*)(bs + (16 + n) * ST1 + kr);   // cols 16..23 (+zero pad)
      acc0 = wmma4(a, b0, acc0);
      acc1 = wmma4(a, b1, acc1);
    }
    __syncthreads();
  }
#pragma unroll
  for (int r = 0; r < 8; ++r) {
    const int row = i0 + r + 8 * half;       // C/D layout: M = r + 8*half, N = n
    float* o = hpart + (size_t)blockIdx.y * (Nn * DMID) + (size_t)row * DMID;
    o[n] = acc0[r];
    if (n < 8) o[16 + n] = acc1[r];
  }
}

// ---------- h = relu(sum_c hpart[c]) ----------
__global__ __launch_bounds__(256) void k_reduce_h(const float* __restrict__ hp,
                                                  float* __restrict__ h) {
  int t = blockIdx.x * 256 + threadIdx.x;
  if (t >= Nn * DMID) return;
  float s = hp[t] + hp[(size_t)Nn * DMID + t] + hp[(size_t)2 * Nn * DMID + t] +
            hp[(size_t)3 * Nn * DMID + t];
  h[t] = fmaxf(s, 0.f);
}

// ---------- T2 = h @ W2  [8192,24]x[24,12] ----------
__global__ __launch_bounds__(256) void k_hw2(const float* __restrict__ h,
                                             const float* __restrict__ W2,
                                             float* __restrict__ T2) {
  int i = blockIdx.x * 256 + threadIdx.x;
  if (i >= Nn) return;
  float acc[DEMB];
#pragma unroll
  for (int n = 0; n < DEMB; ++n) acc[n] = 0.f;
#pragma unroll 4
  for (int k = 0; k < DMID; ++k) {
    float hk = h[(size_t)i * DMID + k];
#pragma unroll
    for (int n = 0; n < DEMB; ++n) acc[n] = fmaf(hk, W2[k * DEMB + n], acc[n]);
  }
#pragma unroll
  for (int n = 0; n < DEMB; ++n) T2[(size_t)i * DEMB + n] = acc[n];
}

// ---------- epart[c] = A-chunk @ T2-chunk  (WMMA f32, 12 cols zero-padded to 16) ----------
__global__ __launch_bounds__(256) void k_gemm2(const float* __restrict__ A,
                                               const float* __restrict__ T2,
                                               float* __restrict__ epart) {
  __shared__ float bs[16 * ST1];             // 8.25 KB; cols 12..15 pre-zeroed
  const int lane = threadIdx.x & 31;
  const int wave = threadIdx.x >> 5;
  const int half = lane >> 4;
  const int n    = lane & 15;
  const int i0   = blockIdx.x * 128 + wave * 16;
  const int kbase = blockIdx.y * KCH;

  for (int t = threadIdx.x; t < 4 * 128; t += 256)
    bs[(12 + (t >> 7)) * ST1 + (t & 127)] = 0.f;
  __syncthreads();

  v8f acc = {0,0,0,0,0,0,0,0};
  const float* aptr = A + (size_t)(i0 + n) * Nn + kbase + 2 * half;

  for (int kt = 0; kt < KCH; kt += 128) {
    for (int t = threadIdx.x; t < 128 * DEMB; t += 256) {
      int row = t / DEMB, col = t - row * DEMB;
      bs[col * ST1 + row] = T2[(size_t)(kbase + kt) * DEMB + t];
    }
    __syncthreads();
#pragma unroll 4
    for (int kk = 0; kk < 128; kk += 4) {
      v2f a = *(const v2f*)(aptr + kt + kk);
      const int kr = kk + 2 * half;
      v2f b = *(const v2f*)(bs + n * ST1 + kr);   // pad cols read 0
      acc = wmma4(a, b, acc);
    }
    __syncthreads();
  }
#pragma unroll
  for (int r = 0; r < 8; ++r) {
    const int row = i0 + r + 8 * half;
    if (n < DEMB)
      epart[(size_t)blockIdx.y * (Nn * DEMB) + (size_t)row * DEMB + n] = acc[r];
  }
}

// ---------- emb = sum_c epart[c] ----------
__global__ __launch_bounds__(256) void k_reduce_emb(const float* __restrict__ ep,
                                                    float* __restrict__ emb) {
  int t = blockIdx.x * 256 + threadIdx.x;
  if (t >= Nn * DEMB) return;
  emb[t] = ep[t] + ep[(size_t)Nn * DEMB + t] + ep[(size_t)2 * Nn * DEMB + t] +
           ep[(size_t)3 * Nn * DEMB + t];
}

// ---------- sq[i] = ||emb[i]||^2 ----------
__global__ __launch_bounds__(256) void k_sq(const float* __restrict__ emb,
                                            float* __restrict__ sq) {
  int i = blockIdx.x * 256 + threadIdx.x;
  if (i >= Nn) return;
  float s = 0.f;
#pragma unroll
  for (int n = 0; n < DEMB; ++n) {
    float e = emb[(size_t)i * DEMB + n];
    s = fmaf(e, e, s);
  }
  sq[i] = s;
}

// ---------- softmax pass 1: per-row partial sums of exp(-relu(dist)) over a j-chunk ----------
// (row max of -dists is exactly 0 via the diagonal, so no max tracking needed)
__global__ __launch_bounds__(256) void k_sm1(const float* __restrict__ emb,
                                             const float* __restrict__ sq,
                                             float* __restrict__ psum) {
  __shared__ float embs[JCH * SE];   // 72 KB shared j-panel (stride 18, conflict-free)
  __shared__ float sqs[JCH];         // 4 KB
  const int lane = threadIdx.x & 31;
  const int wave = threadIdx.x >> 5;
  const int half = lane >> 4;
  const int n    = lane & 15;
  const int i0   = blockIdx.x * 128 + wave * 16;
  const int jbase = blockIdx.y * JCH;

  for (int t = threadIdx.x; t < JCH * DEMB; t += 256) {
    int row = t / DEMB, col = t - row * DEMB;
    embs[row * SE + col] = emb[(size_t)jbase * DEMB + t];
  }
  for (int t = threadIdx.x; t < JCH; t += 256) sqs[t] = sq[jbase + t];
  __syncthreads();

  v2f a0 = *(const v2f*)(emb + (size_t)(i0 + n) * DEMB + 0 + 2 * half);
  v2f a1 = *(const v2f*)(emb + (size_t)(i0 + n) * DEMB + 4 + 2 * half);
  v2f a2 = *(const v2f*)(emb + (size_t)(i0 + n) * DEMB + 8 + 2 * half);

  float si[8], sum[8];
#pragma unroll
  for (int r = 0; r < 8; ++r) { si[r] = sq[i0 + r + 8 * half]; sum[r] = 0.f; }

  for (int jt = 0; jt < JCH; jt += 16) {
    const float* bp = embs + (jt + n) * SE + 2 * half;
    v2f b0 = *(const v2f*)(bp + 0);
    v2f b1 = *(const v2f*)(bp + 4);
    v2f b2 = *(const v2f*)(bp + 8);
    v8f g = {0,0,0,0,0,0,0,0};
    g = wmma4(a0, b0, g);
    g = wmma4(a1, b1, g);
    g = wmma4(a2, b2, g);
    float sj = sqs[jt + n];
#pragma unroll
    for (int r = 0; r < 8; ++r) {
      float d = fmaxf(si[r] + sj - 2.f * g[r], 0.f);
      sum[r] += __expf(-d);
    }
  }
#pragma unroll
  for (int r = 0; r < 8; ++r) {
    float s = sum[r];
#pragma unroll
    for (int m = 1; m < 16; m <<= 1) s += __shfl_xor(s, m, 32);
    if (n == 0)  // lane 0 -> row i0+r, lane 16 -> row i0+8+r
      psum[(size_t)(i0 + r + 8 * half) * (Nn / JCH) + blockIdx.y] = s;
  }
}

// ---------- softmax pass 2: recompute gram, write exp(-d)/rowsum + eps ----------
__global__ __launch_bounds__(256) void k_sm2(const float* __restrict__ emb,
                                             const float* __restrict__ sq,
                                             const float* __restrict__ psum,
                                             float* __restrict__ out) {
  __shared__ float embs[JCH * SE];
  __shared__ float sqs[JCH];
  const int lane = threadIdx.x & 31;
  const int wave = threadIdx.x >> 5;
  const int half = lane >> 4;
  const int n    = lane & 15;
  const int i0   = blockIdx.x * 128 + wave * 16;
  const int jbase = blockIdx.y * JCH;

  for (int t = threadIdx.x; t < JCH * DEMB; t += 256) {
    int row = t / DEMB, col = t - row * DEMB;
    embs[row * SE + col] = emb[(size_t)jbase * DEMB + t];
  }
  for (int t = threadIdx.x; t < JCH; t += 256) sqs[t] = sq[jbase + t];
  __syncthreads();

  v2f a0 = *(const v2f*)(emb + (size_t)(i0 + n) * DEMB + 0 + 2 * half);
  v2f a1 = *(const v2f*)(emb + (size_t)(i0 + n) * DEMB + 4 + 2 * half);
  v2f a2 = *(const v2f*)(emb + (size_t)(i0 + n) * DEMB + 8 + 2 * half);

  float si[8], rinv[8];
#pragma unroll
  for (int r = 0; r < 8; ++r) {
    const int row = i0 + r + 8 * half;
    si[r] = sq[row];
    float tot = 0.f;
#pragma unroll
    for (int c = 0; c < Nn / JCH; ++c) tot += psum[(size_t)row * (Nn / JCH) + c];
    rinv[r] = 1.f / tot;
  }

  for (int jt = 0; jt < JCH; jt += 16) {
    const float* bp = embs + (jt + n) * SE + 2 * half;
    v2f b0 = *(const v2f*)(bp + 0);
    v2f b1 = *(const v2f*)(bp + 4);
    v2f b2 = *(const v2f*)(bp + 8);
    v8f g = {0,0,0,0,0,0,0,0};
    g = wmma4(a0, b0, g);
    g = wmma4(a1, b1, g);
    g = wmma4(a2, b2, g);
    float sj = sqs[jt + n];
#pragma unroll
    for (int r = 0; r < 8; ++r) {
      float d = fmaxf(si[r] + sj - 2.f * g[r], 0.f);
      out[(size_t)(i0 + r + 8 * half) * Nn + jbase + jt + n] = __expf(-d) * rinv[r] + 1e-10f;
    }
  }
}

extern "C" void kernel_launch(void* const* d_in, const int* in_sizes, int n_in,
                              void* d_out, int out_size, void* d_ws, size_t ws_size,
                              hipStream_t stream) {
  const float* A  = (const float*)d_in[0];   // [8192,8192]
  const float* X  = (const float*)d_in[1];   // [8192,32]
  const float* W1 = (const float*)d_in[2];   // [32,24]
  const float* W2 = (const float*)d_in[3];   // [24,12]
  float* out = (float*)d_out;
  float* ws  = (float*)d_ws;

  float* T1    = ws;                              // 8192*24
  float* hpart = T1 + (size_t)Nn * DMID;          // 4 * 8192*24
  float* h     = hpart + (size_t)4 * Nn * DMID;   // 8192*24
  float* T2    = h + (size_t)Nn * DMID;           // 8192*12
  float* epart = T2 + (size_t)Nn * DEMB;          // 4 * 8192*12
  float* emb   = epart + (size_t)4 * Nn * DEMB;   // 8192*12
  float* sq    = emb + (size_t)Nn * DEMB;         // 8192
  float* psum  = sq + Nn;                         // 8192*8  (~7 MB total)

  k_xw1<<<Nn / 256, 256, 0, stream>>>(X, W1, T1);
  k_gemm1<<<dim3(Nn / 128, 4), 256, 0, stream>>>(A, T1, hpart);
  k_reduce_h<<<(Nn * DMID) / 256, 256, 0, stream>>>(hpart, h);
  k_hw2<<<Nn / 256, 256, 0, stream>>>(h, W2, T2);
  k_gemm2<<<dim3(Nn / 128, 4), 256, 0, stream>>>(A, T2, epart);
  k_reduce_emb<<<(Nn * DEMB) / 256, 256, 0, stream>>>(epart, emb);
  k_sq<<<Nn / 256, 256, 0, stream>>>(emb, sq);
  k_sm1<<<dim3(Nn / 128, Nn / JCH), 256, 0, stream>>>(emb, sq, psum);
  k_sm2<<<dim3(Nn / 128, Nn / JCH), 256, 0, stream>>>(emb, sq, psum, out);
}